// GNN_34686155882550
// MI455X (gfx1250) — compile-verified
//
#include <hip/hip_runtime.h>

#define N_ 50000
#define E_ 800000
#define D_ 64
#define G_ 512
#define KTOT 128          // combined K = [agg | h]
#define APAD 132          // padded LDS row stride for A tile (bank-conflict free)

typedef __attribute__((ext_vector_type(2))) float v2f;
typedef __attribute__((ext_vector_type(8))) float v8f;

// ---------------------------------------------------------------------------
// Edge scatter: agg[dst] += h[src].  16 threads per edge, float4 per thread,
// native f32 atomics (relaxed, agent scope -> global_atomic_add_f32, L2-resident).
// ---------------------------------------------------------------------------
__global__ __launch_bounds__(256) void scatter_kernel(
    const float* __restrict__ h, const int* __restrict__ src,
    const int* __restrict__ dst, float* __restrict__ agg, int nEdges)
{
    int gid = blockIdx.x * 256 + threadIdx.x;
    int e = gid >> 4;
    if (e >= nEdges) return;
    int cg = (gid & 15) << 2;                     // channel group of 4
    int s = src[e];
    int d = dst[e];
    const float4 v = *reinterpret_cast<const float4*>(h + (size_t)s * D_ + cg);
    float* p = agg + (size_t)d * D_ + cg;
    __hip_atomic_fetch_add(p + 0, v.x, __ATOMIC_RELAXED, __HIP_MEMORY_SCOPE_AGENT);
    __hip_atomic_fetch_add(p + 1, v.y, __ATOMIC_RELAXED, __HIP_MEMORY_SCOPE_AGENT);
    __hip_atomic_fetch_add(p + 2, v.z, __ATOMIC_RELAXED, __HIP_MEMORY_SCOPE_AGENT);
    __hip_atomic_fetch_add(p + 3, v.w, __ATOMIC_RELAXED, __HIP_MEMORY_SCOPE_AGENT);
}

// ---------------------------------------------------------------------------
// Fused GraphConv GEMM via V_WMMA_F32_16X16X4_F32:
//   out = relu( [residual +] agg @ Wrel^T + b + h @ Wroot^T )
// One block = 16 output rows, 4 waves x one 16-col tile each, K=128 combined.
// ---------------------------------------------------------------------------
__global__ __launch_bounds__(128) void gemm_kernel(
    const float* __restrict__ agg, const float* __restrict__ h,
    const float* __restrict__ wrel, const float* __restrict__ wroot,
    const float* __restrict__ brel, float* __restrict__ out, int residual)
{
    __shared__ float wlds[KTOT * D_];     // B: [k][d], Wc[k][d]
    __shared__ float alds[16 * APAD];     // A: [row][k], padded stride

    const int tid  = threadIdx.x;
    const int lane = tid & 31;
    const int wave = tid >> 5;            // 0..3 -> 16-col tile
    const int r0   = blockIdx.x * 16;

    // Stage combined weights: Wc[k][d] = k<64 ? wrel[d][k] : wroot[d][k-64]
    for (int i = tid; i < KTOT * D_; i += 128) {
        int k = i >> 6;
        int d = i & 63;
        wlds[i] = (k < D_) ? wrel[d * D_ + k] : wroot[d * D_ + (k - D_)];
    }
    // Stage A tile: [agg | h] rows r0..r0+15, coalesced along k
    for (int i = tid; i < 16 * KTOT; i += 128) {
        int row = i >> 7;
        int k   = i & 127;
        float v = (k < D_) ? agg[(size_t)(r0 + row) * D_ + k]
                           : h[(size_t)(r0 + row) * D_ + (k - D_)];
        alds[row * APAD + k] = v;
    }
    __syncthreads();

    const int c0   = wave * 16;
    const int lrow = lane & 15;
    const int khi  = (lane >> 4) << 1;    // lanes 0-15 -> K{0,1}; 16-31 -> K{2,3}

    v8f acc = {0.f, 0.f, 0.f, 0.f, 0.f, 0.f, 0.f, 0.f};
#pragma unroll
    for (int kb = 0; kb < KTOT; kb += 4) {
        v2f a, b;
        a.x = alds[lrow * APAD + kb + khi + 0];
        a.y = alds[lrow * APAD + kb + khi + 1];
        b.x = wlds[(kb + khi + 0) * D_ + c0 + lrow];
        b.y = wlds[(kb + khi + 1) * D_ + c0 + lrow];
        acc = __builtin_amdgcn_wmma_f32_16x16x4_f32(
            /*neg_a=*/false, a, /*neg_b=*/false, b,
            /*c_mod=*/(short)0, acc, /*reuse_a=*/false, /*reuse_b=*/false);
    }

    // Epilogue: bias + optional residual + ReLU.
    // C layout: VGPR v holds row v (lanes 0-15) and row v+8 (lanes 16-31), col = lane%16.
    const int col   = c0 + lrow;
    const float bias = brel[col];
    const int rbase = r0 + ((lane >> 4) << 3);
#pragma unroll
    for (int v = 0; v < 8; ++v) {
        int row = rbase + v;
        float val = acc[v] + bias;
        if (residual) val += h[(size_t)row * D_ + col];
        out[(size_t)row * D_ + col] = fmaxf(val, 0.f);
    }
}

// ---------------------------------------------------------------------------
// Mean-pool accumulation: sums[batch[n]] += h[n], cnts[batch[n]] += 1
// ---------------------------------------------------------------------------
__global__ __launch_bounds__(256) void pool_kernel(
    const float* __restrict__ h, const int* __restrict__ batch,
    float* __restrict__ sums, float* __restrict__ cnts)
{
    int gid = blockIdx.x * 256 + threadIdx.x;
    int n = gid >> 4;
    if (n >= N_) return;
    int cg = (gid & 15) << 2;
    int g = batch[n];
    const float4 v = *reinterpret_cast<const float4*>(h + (size_t)n * D_ + cg);
    float* p = sums + (size_t)g * D_ + cg;
    __hip_atomic_fetch_add(p + 0, v.x, __ATOMIC_RELAXED, __HIP_MEMORY_SCOPE_AGENT);
    __hip_atomic_fetch_add(p + 1, v.y, __ATOMIC_RELAXED, __HIP_MEMORY_SCOPE_AGENT);
    __hip_atomic_fetch_add(p + 2, v.z, __ATOMIC_RELAXED, __HIP_MEMORY_SCOPE_AGENT);
    __hip_atomic_fetch_add(p + 3, v.w, __ATOMIC_RELAXED, __HIP_MEMORY_SCOPE_AGENT);
    if ((gid & 15) == 0)
        __hip_atomic_fetch_add(cnts + g, 1.0f, __ATOMIC_RELAXED, __HIP_MEMORY_SCOPE_AGENT);
}

// ---------------------------------------------------------------------------
// Head: out[g] = softmax( (sums[g]/max(cnt,1)) @ lin_w^T + lin_b )
// One 64-thread block per graph; LDS tree reductions for max / sum.
// ---------------------------------------------------------------------------
__global__ __launch_bounds__(64) void head_kernel(
    const float* __restrict__ sums, const float* __restrict__ cnts,
    const float* __restrict__ lw, const float* __restrict__ lb,
    float* __restrict__ out)
{
    __shared__ float pl[D_];
    __shared__ float red[D_];
    int g = blockIdx.x;
    int d = threadIdx.x;
    float c = fmaxf(cnts[g], 1.0f);
    pl[d] = sums[(size_t)g * D_ + d] / c;
    __syncthreads();

    float z = lb[d];
#pragma unroll
    for (int k = 0; k < D_; ++k) z += pl[k] * lw[d * D_ + k];

    red[d] = z;
    __syncthreads();
    for (int s = 32; s > 0; s >>= 1) {
        if (d < s) red[d] = fmaxf(red[d], red[d + s]);
        __syncthreads();
    }
    float m = red[0];
    __syncthreads();
    float e = __expf(z - m);
    red[d] = e;
    __syncthreads();
    for (int s = 32; s > 0; s >>= 1) {
        if (d < s) red[d] += red[d + s];
        __syncthreads();
    }
    out[(size_t)g * D_ + d] = e / red[0];
}

// ---------------------------------------------------------------------------
extern "C" void kernel_launch(void* const* d_in, const int* in_sizes, int n_in,
                              void* d_out, int out_size, void* d_ws, size_t ws_size,
                              hipStream_t stream)
{
    const float* x      = (const float*)d_in[0];
    const int*   ei     = (const int*)d_in[1];
    const int*   batch  = (const int*)d_in[2];
    const float* rel_w  = (const float*)d_in[3];
    const float* rel_b  = (const float*)d_in[4];
    const float* root_w = (const float*)d_in[5];
    const float* lin_w  = (const float*)d_in[6];
    const float* lin_b  = (const float*)d_in[7];
    float* out = (float*)d_out;

    const int* srcIdx = ei;
    const int* dstIdx = ei + E_;

    const size_t ND = (size_t)N_ * D_;
    float* ws   = (float*)d_ws;
    float* bufA = ws;
    float* bufB = ws + ND;
    float* agg  = ws + 2 * ND;
    float* sums = ws + 3 * ND;
    float* cnts = sums + (size_t)G_ * D_;

    const float* cur = x;
    float* bufs[2] = {bufA, bufB};
    for (int l = 0; l < 5; ++l) {
        float* nxt = bufs[l & 1];
        hipMemsetAsync(agg, 0, ND * sizeof(float), stream);
        scatter_kernel<<<(E_ * 16) / 256, 256, 0, stream>>>(cur, srcIdx, dstIdx, agg, E_);
        gemm_kernel<<<N_ / 16, 128, 0, stream>>>(
            agg, cur, rel_w + (size_t)l * D_ * D_, root_w + (size_t)l * D_ * D_,
            rel_b + (size_t)l * D_, nxt, (l >= 3) ? 1 : 0);
        cur = nxt;
    }

    hipMemsetAsync(sums, 0, ((size_t)G_ * D_ + G_) * sizeof(float), stream);
    pool_kernel<<<(N_ * 16 + 255) / 256, 256, 0, stream>>>(cur, batch, sums, cnts);
    head_kernel<<<G_, 64, 0, stream>>>(sums, cnts, lin_w, lin_b, out);
}